// Whiten_71734543777922
// MI455X (gfx1250) — compile-verified
//
#include <hip/hip_runtime.h>

// Whitening: per-(b,c) mean/std(ddof=1) over 224x224, std clamped at 1/224.
// B=64, C=32, H=W=224  ->  2048 slices of 50176 fp32 (200704 B each).
// MI455X plan: one workgroup per slice; stage the whole slice into the
// 320KB-per-WGP LDS via CDNA5 async-to-LDS (ASYNCcnt) loads; reduction pass
// is software-pipelined against the in-flight async loads using staged
// s_wait_asynccnt (loads retire in order per wave, and each thread consumes
// exactly the LDS chunks it staged, so no cross-wave barrier is needed before
// the stats pass). HBM traffic = 1R + 1W (~822 MB @ 23.3 TB/s => ~35 us).

#define NSLICE   2048
#define HW       50176
#define THREADS  256
#define VEC_ITERS 49   // (HW/4) / THREADS == 12544 / 256

typedef float float4v __attribute__((ext_vector_type(4)));
typedef int   v4i     __attribute__((ext_vector_type(4)));
typedef __attribute__((address_space(1))) v4i  gv4i;   // HIP: __device__ (AS1)
typedef __attribute__((address_space(3))) v4i  lv4i;   // HIP: __shared__ (AS3)
typedef __attribute__((address_space(3))) void as3_void;

__device__ __forceinline__ void async_copy16(const float* gsrc, float* ldst) {
#if __has_builtin(__builtin_amdgcn_global_load_async_to_lds_b128)
  __builtin_amdgcn_global_load_async_to_lds_b128(
      (gv4i*)(unsigned long long)(const void*)gsrc,   // inttoptr -> AS1 v4i*
      (lv4i*)(as3_void*)ldst,                         // addrspacecast -> AS3
      /*imm offset*/ 0, /*cpol*/ 0);
#else
  unsigned lds_off = (unsigned)(unsigned long long)(lv4i*)(as3_void*)ldst;
  unsigned long long ga = (unsigned long long)(const void*)gsrc;
  asm volatile("global_load_async_to_lds_b128 %0, %1, off"
               :: "v"(lds_off), "v"(ga) : "memory");
#endif
}

template <int N>
__device__ __forceinline__ void wait_async_le() {
#if __has_builtin(__builtin_amdgcn_s_wait_asynccnt)
  __builtin_amdgcn_s_wait_asynccnt(N);
#else
  asm volatile("s_wait_asynccnt %0" :: "i"(N) : "memory");
#endif
}

// Pipelined stats pass: wait until this wave's first (I+1) async loads have
// retired (ASYNCcnt <= VEC_ITERS-1-I), then accumulate chunk I from LDS.
template <int I>
__device__ __forceinline__ void stats_chunks(const float4v* smem4, int tid,
                                             float& sum, float& sq) {
  if constexpr (I < VEC_ITERS) {
    wait_async_le<VEC_ITERS - 1 - I>();
    float4v v = smem4[tid + I * THREADS];
    sum += (v.x + v.y) + (v.z + v.w);
    sq = fmaf(v.x, v.x, sq);
    sq = fmaf(v.y, v.y, sq);
    sq = fmaf(v.z, v.z, sq);
    sq = fmaf(v.w, v.w, sq);
    stats_chunks<I + 1>(smem4, tid, sum, sq);
  }
}

__global__ __launch_bounds__(THREADS)
void whiten_kernel(const float* __restrict__ x, float* __restrict__ y) {
  extern __shared__ float4v smem4[];           // 12544 float4 = 200704 B
  float* smem = reinterpret_cast<float*>(smem4);
  __shared__ float wsum[THREADS / 32];
  __shared__ float wsq[THREADS / 32];
  __shared__ float s_mean;
  __shared__ float s_rstd;

  const int tid = threadIdx.x;
  const long long base = (long long)blockIdx.x * (long long)HW;
  const float* gsrc = x + base;
  float* gdst = y + base;

  // ---- Stage slice HBM -> LDS: 49 x async b128 per lane (ASYNCcnt <= 49) ----
  #pragma unroll
  for (int i = 0; i < VEC_ITERS; ++i) {
    const int v = tid + i * THREADS;           // float4 index within slice
    async_copy16(gsrc + 4 * v, smem + 4 * v);
  }

  // ---- Pass 1 (pipelined against async loads): sum / sum-of-squares ----
  float sum = 0.f, sq = 0.f;
  stats_chunks<0>(smem4, tid, sum, sq);        // ends with s_wait_asynccnt 0

  // wave32 tree reduce
  #pragma unroll
  for (int off = 16; off > 0; off >>= 1) {
    sum += __shfl_xor(sum, off, 32);
    sq  += __shfl_xor(sq,  off, 32);
  }
  if ((tid & 31) == 0) {
    wsum[tid >> 5] = sum;
    wsq[tid >> 5]  = sq;
  }
  __syncthreads();

  if (tid == 0) {
    double ts = 0.0, tq = 0.0;
    #pragma unroll
    for (int w = 0; w < THREADS / 32; ++w) { ts += (double)wsum[w]; tq += (double)wsq[w]; }
    const double n = (double)HW;
    const double mean = ts / n;
    double var = (tq - ts * ts / n) / (n - 1.0);   // unbiased, ddof=1
    if (var < 0.0) var = 0.0;
    double dev = sqrt(var);
    const double mindev = 1.0 / 224.0;             // 1/sqrt(H*W)
    if (dev < mindev) dev = mindev;
    s_mean = (float)mean;
    s_rstd = (float)(1.0 / dev);
  }
  __syncthreads();

  // ---- Pass 2: normalize from LDS (thread-private chunks), NT store ----
  const float m = s_mean;
  const float r = s_rstd;
  float4v* __restrict__ ov = reinterpret_cast<float4v*>(gdst);
  #pragma unroll
  for (int i = 0; i < VEC_ITERS; ++i) {
    const int vi = tid + i * THREADS;
    float4v v = smem4[vi];
    float4v o;
    o.x = (v.x - m) * r;
    o.y = (v.y - m) * r;
    o.z = (v.z - m) * r;
    o.w = (v.w - m) * r;
    __builtin_nontemporal_store(o, ov + vi);
  }
}

extern "C" void kernel_launch(void* const* d_in, const int* in_sizes, int n_in,
                              void* d_out, int out_size, void* d_ws, size_t ws_size,
                              hipStream_t stream) {
  (void)in_sizes; (void)n_in; (void)d_ws; (void)ws_size; (void)out_size;
  const float* x = (const float*)d_in[0];
  float* y = (float*)d_out;
  const size_t shmem = (size_t)HW * sizeof(float);   // 200704 B < 320 KB/WGP
  (void)hipFuncSetAttribute((const void*)whiten_kernel,
                            hipFuncAttributeMaxDynamicSharedMemorySize, (int)shmem);
  whiten_kernel<<<NSLICE, THREADS, shmem, stream>>>(x, y);
}